// _LearnableHashGrid_33397665693995
// MI455X (gfx1250) — compile-verified
//
#include <hip/hip_runtime.h>
#include <stdint.h>

// Learnable hash grid (instant-NGP style) for MI455X / gfx1250.
// L2-gather bound workload: tables (32 MB) fit in 192 MB L2; no matmul
// structure -> WMMA inapplicable. Uses wave32 thread-per-point, 128-bit
// gathers, v_exp/v_rcp, and gfx1250 async global->LDS staging of x.

#define HG_NF   4
#define HG_NL   8
#define HG_FTS  (1u << 20)   // FEATURE_TABLE_SIZE
#define HG_ITS  (1u << 19)   // INDEX_TABLE_SIZE
#define HG_RES  128.0f
#define HG_P1   2654435761u  // PRIMES[1]
#define HG_P2   805459861u   // PRIMES[2]
#define HG_BLK  256

#if defined(__has_builtin)
#if __has_builtin(__builtin_amdgcn_global_load_async_to_lds_b32)
#define HG_ASYNC 1
#endif
#endif
#ifndef HG_ASYNC
#define HG_ASYNC 0
#endif

__global__ __launch_bounds__(HG_BLK)
void _LearnableHashGrid_33397665693995_kernel(const float* __restrict__ x,
                                              const float* __restrict__ ftab,
                                              const float* __restrict__ itab,
                                              float* __restrict__ out,
                                              int n)
{
    const int tid = blockIdx.x * blockDim.x + threadIdx.x;
    const int pc  = (tid < n) ? tid : (n - 1);   // clamp so every lane stays in-bounds

    float xs0, xs1, xs2;
#if HG_ASYNC
    __shared__ float xt[HG_BLK * 3];
    {
        // gfx1250 async global->LDS copy of this lane's 3 input floats.
        // Builtin params are address-space-qualified int* (AS1 global src,
        // AS3 LDS dst). AS pointers built via integer round-trip: generic-LDS
        // low 32 bits are the DS byte offset (ISA aperture rules).
        float* gp = const_cast<float*>(x) + (size_t)pc * 3u;
        float* lp = &xt[threadIdx.x * 3u];
        typedef __attribute__((address_space(1))) int gi_t;
        typedef __attribute__((address_space(3))) int li_t;
        gi_t* g = (gi_t*)(unsigned long long)(uintptr_t)gp;
        li_t* l = (li_t*)(unsigned int)(uintptr_t)lp;
        __builtin_amdgcn_global_load_async_to_lds_b32(g, l, 0, 0);
        __builtin_amdgcn_global_load_async_to_lds_b32(g, l, 4, 0);
        __builtin_amdgcn_global_load_async_to_lds_b32(g, l, 8, 0);
#if __has_builtin(__builtin_amdgcn_s_wait_asynccnt)
        __builtin_amdgcn_s_wait_asynccnt(0);
#else
        asm volatile("s_wait_asynccnt 0" ::: "memory");
#endif
        xs0 = xt[threadIdx.x * 3u + 0] * HG_RES;
        xs1 = xt[threadIdx.x * 3u + 1] * HG_RES;
        xs2 = xt[threadIdx.x * 3u + 2] * HG_RES;
    }
#else
    {
        const float* xp = x + (size_t)pc * 3u;
        xs0 = xp[0] * HG_RES;
        xs1 = xp[1] * HG_RES;
        xs2 = xp[2] * HG_RES;
    }
#endif

    const int xi0 = (int)xs0, xi1 = (int)xs1, xi2 = (int)xs2;
    const float xf0 = xs0 - (float)xi0;
    const float xf1 = xs1 - (float)xi1;
    const float xf2 = xs2 - (float)xi2;

    // hash partials: corner bit set -> (xi_d+1)*P_d = xi_d*P_d + P_d
    const unsigned A0 = (unsigned)xi0;            // PRIMES[0] == 1
    const unsigned A1 = (unsigned)xi1 * HG_P1;
    const unsigned A2 = (unsigned)xi2 * HG_P2;
    const unsigned B0 = A0 + 1u;
    const unsigned B1 = A1 + HG_P1;
    const unsigned B2 = A2 + HG_P2;

    float o0 = 0.f, o1 = 0.f, o2 = 0.f, o3 = 0.f;

#pragma unroll
    for (int c = 0; c < 8; ++c) {
        unsigned h = ((c & 1) ? B0 : A0) ^ ((c & 2) ? B1 : A1) ^ ((c & 4) ? B2 : A2);
        h &= (HG_ITS - 1u);

        const float wc = ((c & 1) ? xf0 : 1.0f - xf0)
                       * ((c & 2) ? xf1 : 1.0f - xf1)
                       * ((c & 4) ? xf2 : 1.0f - xf2);

        // index_table row: 8 floats, 32B aligned -> two b128 gathers
        const float4* iwp = (const float4*)(itab + (size_t)h * HG_NL);
        const float4 iw0 = iwp[0];
        const float4 iw1 = iwp[1];

        // feature_table addresses: a = (l*P1 ^ h*P2) & (2^20-1); l*P1 const-folds
        const unsigned hb = h * HG_P2;
        unsigned a[HG_NL];
#pragma unroll
        for (int l = 0; l < HG_NL; ++l)
            a[l] = (((unsigned)l * HG_P1) ^ hb) & (HG_FTS - 1u);

        // issue all 8 b128 gathers before consuming (one loadcnt wait batch)
        float4 f[HG_NL];
#pragma unroll
        for (int l = 0; l < HG_NL; ++l)
            f[l] = *(const float4*)(ftab + (size_t)a[l] * HG_NF);

        // softmax over the 8 learnable weights (max-subtracted, v_exp_f32)
        const float m = fmaxf(fmaxf(fmaxf(iw0.x, iw0.y), fmaxf(iw0.z, iw0.w)),
                              fmaxf(fmaxf(iw1.x, iw1.y), fmaxf(iw1.z, iw1.w)));
        float e[HG_NL];
        e[0] = __expf(iw0.x - m); e[1] = __expf(iw0.y - m);
        e[2] = __expf(iw0.z - m); e[3] = __expf(iw0.w - m);
        e[4] = __expf(iw1.x - m); e[5] = __expf(iw1.y - m);
        e[6] = __expf(iw1.z - m); e[7] = __expf(iw1.w - m);
        const float s = ((e[0] + e[1]) + (e[2] + e[3])) + ((e[4] + e[5]) + (e[6] + e[7]));

        float c0 = 0.f, c1 = 0.f, c2 = 0.f, c3 = 0.f;
#pragma unroll
        for (int l = 0; l < HG_NL; ++l) {
            c0 = fmaf(e[l], f[l].x, c0);
            c1 = fmaf(e[l], f[l].y, c1);
            c2 = fmaf(e[l], f[l].z, c2);
            c3 = fmaf(e[l], f[l].w, c3);
        }

        // fold softmax normalization into the trilinear corner weight
        const float scale = wc * __builtin_amdgcn_rcpf(s);
        o0 = fmaf(scale, c0, o0);
        o1 = fmaf(scale, c1, o1);
        o2 = fmaf(scale, c2, o2);
        o3 = fmaf(scale, c3, o3);
    }

    if (tid < n) {
        float4 r; r.x = o0; r.y = o1; r.z = o2; r.w = o3;
        *(float4*)(out + (size_t)tid * 4u) = r;   // single b128 store
    }
}

extern "C" void kernel_launch(void* const* d_in, const int* in_sizes, int n_in,
                              void* d_out, int out_size, void* d_ws, size_t ws_size,
                              hipStream_t stream) {
    (void)n_in; (void)out_size; (void)d_ws; (void)ws_size;
    const float* x    = (const float*)d_in[0];   // (N_POINTS, 3) f32
    const float* ftab = (const float*)d_in[1];   // (2^20, 4)    f32
    const float* itab = (const float*)d_in[2];   // (2^19, 8)    f32
    float* out = (float*)d_out;                  // (N_POINTS, 4) f32

    const int n = in_sizes[0] / 3;
    if (n <= 0) return;
    const int blocks = (n + HG_BLK - 1) / HG_BLK;
    _LearnableHashGrid_33397665693995_kernel<<<dim3(blocks), dim3(HG_BLK), 0, stream>>>(
        x, ftab, itab, out, n);
}